// RGCN_classify_34385508171923
// MI455X (gfx1250) — compile-verified
//
#include <hip/hip_runtime.h>
#include <hip/hip_bf16.h>

#define N_NODES 100000
#define N_EDGES 1600000
#define N_REL   4
#define SLOPE   0.2f

typedef __attribute__((ext_vector_type(2))) float v2f;
typedef __attribute__((ext_vector_type(8))) float v8f;

__device__ __forceinline__ float leaky_f(float x) { return fmaxf(x, SLOPE * x); }

// ---------------------------------------------------------------- utilities
__global__ void zero_kernel(float* __restrict__ p, int n) {
  int i = blockIdx.x * blockDim.x + threadIdx.x;
  if (i < n) p[i] = 0.0f;
}

// count in/out degrees for all relations at once (src/dst are [R*E] flat)
__global__ void degree_kernel(const int* __restrict__ src, const int* __restrict__ dst,
                              float* __restrict__ deg_out, float* __restrict__ deg_in) {
  int i = blockIdx.x * blockDim.x + threadIdx.x;
  if (i < N_REL * N_EDGES) {
    int r = i / N_EDGES;
    atomicAdd(&deg_out[r * N_NODES + src[i]], 1.0f);
    atomicAdd(&deg_in [r * N_NODES + dst[i]], 1.0f);
  }
}

// deg -> deg>0 ? 1/sqrt(deg) : 0   (in place, covers c_src and c_dst regions)
__global__ void norm_kernel(float* __restrict__ deg, int n) {
  int i = blockIdx.x * blockDim.x + threadIdx.x;
  if (i < n) {
    float d = deg[i];
    deg[i] = (d > 0.0f) ? rsqrtf(d) : 0.0f;
  }
}

// ------------------------------------------------------- WMMA fp32 GEMM
// h[m0..m0+15][*] = (pre_act? leaky(x) : x) * c_src[m]  @  W   (DIN x DOUT)
// one block = 16-node M tile; wave w owns output columns [16w, 16w+16)
template <int DIN, int DOUT>
__global__ __launch_bounds__(32 * (DOUT / 16))
void gemm_wmma(const float* __restrict__ x, const float* __restrict__ c_src,
               const float* __restrict__ W, float* __restrict__ h, int pre_act) {
  constexpr int NTHREADS = 32 * (DOUT / 16);
  constexpr int LDA = DIN + 4;               // +4 floats: conflict-free banks
  __shared__ float tileA[16 * LDA];

  const int m0  = blockIdx.x * 16;
  const int tid = threadIdx.x;

  // stage 16 x DIN input tile, pre-scaled by c_src (and external leaky if set)
  for (int idx = tid; idx < 16 * DIN; idx += NTHREADS) {
    int row = idx / DIN, col = idx % DIN;
    float v = x[(size_t)(m0 + row) * DIN + col];
    if (pre_act) v = fmaxf(v, SLOPE * v);
    tileA[row * LDA + col] = v * c_src[m0 + row];
  }
  __syncthreads();

  const int wave = tid >> 5;
  const int lane = tid & 31;
  const int n0   = wave * 16;
  const int m    = lane & 15;          // A row this lane feeds
  const int kb   = (lane >> 4) * 2;    // K sub-offset per half-wave
  const int nl   = lane & 15;          // B/C column this lane feeds

  v8f acc = {};
  for (int k0 = 0; k0 < DIN; k0 += 4) {
    v2f a, b;
    a.x = tileA[m * LDA + (k0 + kb)];
    a.y = tileA[m * LDA + (k0 + kb + 1)];
    b.x = W[(size_t)(k0 + kb)     * DOUT + n0 + nl];
    b.y = W[(size_t)(k0 + kb + 1) * DOUT + n0 + nl];
    acc = __builtin_amdgcn_wmma_f32_16x16x4_f32(
        /*neg_a=*/false, a, /*neg_b=*/false, b,
        /*c_mod=*/(short)0, acc, /*reuse_a=*/false, /*reuse_b=*/false);
  }

  // C/D layout: VGPR v -> M = v + 8*(lane>=16), N = lane%16
  const int rbase = m0 + ((lane >> 4) << 3);
#pragma unroll
  for (int v = 0; v < 8; ++v)
    h[(size_t)(rbase + v) * DOUT + n0 + nl] = acc[v];
}

// ------------------------------------------------- edge gather / scatter-add
template <int DOUT>
__global__ void scatter_kernel(const float* __restrict__ h, const int* __restrict__ src,
                               const int* __restrict__ dst, float* __restrict__ agg) {
  constexpr int CHUNK = DOUT / 4;            // float4 per thread
  int i = blockIdx.x * blockDim.x + threadIdx.x;
  if (i >= N_EDGES * CHUNK) return;
  int e = i / CHUNK;
  int f = (i % CHUNK) * 4;
  int s = src[e], d = dst[e];
  float4 v = *(const float4*)(h + (size_t)s * DOUT + f);
  float* o = agg + (size_t)d * DOUT + f;
  atomicAdd(o + 0, v.x);
  atomicAdd(o + 1, v.y);
  atomicAdd(o + 2, v.z);
  atomicAdd(o + 3, v.w);
}

// acc += leaky(agg * c_dst + b)   (per-relation activation + relation sum)
template <int DOUT>
__global__ void combine_kernel(const float* __restrict__ agg, const float* __restrict__ c_dst,
                               const float* __restrict__ b, float* __restrict__ acc) {
  int i = blockIdx.x * blockDim.x + threadIdx.x;
  if (i >= N_NODES * DOUT) return;
  int n = i / DOUT, f = i % DOUT;
  float v = agg[i] * c_dst[n] + b[f];
  acc[i] += fmaxf(v, SLOPE * v);
}

// ---------------------------------------------------------------- layer driver
template <int DIN, int DOUT>
static void run_layer(const float* xin, float* acc_out,
                      const float* Wall, const float* ball,
                      const float* c_src, const float* c_dst,
                      const int* esrc, const int* edst,
                      float* hbuf, float* agg, int pre_act, hipStream_t stream) {
  const int outN = N_NODES * DOUT;
  zero_kernel<<<(outN + 255) / 256, 256, 0, stream>>>(acc_out, outN);
  for (int r = 0; r < N_REL; ++r) {
    zero_kernel<<<(outN + 255) / 256, 256, 0, stream>>>(agg, outN);
    gemm_wmma<DIN, DOUT><<<N_NODES / 16, 32 * (DOUT / 16), 0, stream>>>(
        xin, c_src + (size_t)r * N_NODES, Wall + (size_t)r * DIN * DOUT, hbuf, pre_act);
    int swork = N_EDGES * (DOUT / 4);
    scatter_kernel<DOUT><<<(swork + 255) / 256, 256, 0, stream>>>(
        hbuf, esrc + (size_t)r * N_EDGES, edst + (size_t)r * N_EDGES, agg);
    combine_kernel<DOUT><<<(outN + 255) / 256, 256, 0, stream>>>(
        agg, c_dst + (size_t)r * N_NODES, ball + (size_t)r * DOUT, acc_out);
  }
}

extern "C" void kernel_launch(void* const* d_in, const int* in_sizes, int n_in,
                              void* d_out, int out_size, void* d_ws, size_t ws_size,
                              hipStream_t stream) {
  (void)in_sizes; (void)n_in; (void)out_size; (void)ws_size;
  const float* x    = (const float*)d_in[0];
  const int*   esrc = (const int*)  d_in[1];
  const int*   edst = (const int*)  d_in[2];
  const float* W1   = (const float*)d_in[3];
  const float* b1   = (const float*)d_in[4];
  const float* W2   = (const float*)d_in[5];
  const float* b2   = (const float*)d_in[6];
  const float* W3   = (const float*)d_in[7];
  const float* b3   = (const float*)d_in[8];
  float* out = (float*)d_out;

  // workspace layout (floats)
  float* ws    = (float*)d_ws;
  float* c_src = ws;                                   // R*N
  float* c_dst = c_src + (size_t)N_REL * N_NODES;      // R*N
  float* hbuf  = c_dst + (size_t)N_REL * N_NODES;      // N*128 (max DOUT)
  float* agg   = hbuf  + (size_t)N_NODES * 128;        // N*128
  float* acc1  = agg   + (size_t)N_NODES * 128;        // N*128
  float* acc2  = acc1  + (size_t)N_NODES * 128;        // N*64

  // ---- degrees -> symmetric norms (shared by all layers)
  int degN = 2 * N_REL * N_NODES;
  zero_kernel<<<(degN + 255) / 256, 256, 0, stream>>>(c_src, degN);
  int ewN = N_REL * N_EDGES;
  degree_kernel<<<(ewN + 255) / 256, 256, 0, stream>>>(esrc, edst, c_src, c_dst);
  norm_kernel<<<(degN + 255) / 256, 256, 0, stream>>>(c_src, degN);

  // ---- 3 hetero layers; external leaky folded into next layer's GEMM staging
  run_layer<256, 128>(x,    acc1, W1, b1, c_src, c_dst, esrc, edst, hbuf, agg, 0, stream);
  run_layer<128,  64>(acc1, acc2, W2, b2, c_src, c_dst, esrc, edst, hbuf, agg, 1, stream);
  run_layer< 64,  32>(acc2, out,  W3, b3, c_src, c_dst, esrc, edst, hbuf, agg, 1, stream);
}